// TrafficGCN_25649544692374
// MI455X (gfx1250) — compile-verified
//
#include <hip/hip_runtime.h>

#define N_NODES 50000
#define INC 3
#define HID 128
#define OUTC 64

typedef __attribute__((ext_vector_type(2))) float v2f;
typedef __attribute__((ext_vector_type(8))) float v8f;

// ---------------- init: zero accumulators, deg = 1 (self loop) ----------------
__global__ void gcn_init(float* __restrict__ agg1, float* __restrict__ out,
                         float* __restrict__ deg) {
    int i = blockIdx.x * blockDim.x + threadIdx.x;
    if (i < N_NODES * HID)  agg1[i] = 0.0f;
    if (i < N_NODES * OUTC) out[i]  = 0.0f;
    if (i < N_NODES)        deg[i]  = 1.0f;   // self-loop contribution
}

// ---------------- degree count over edges ----------------
__global__ void gcn_degree(const int* __restrict__ dst, float* __restrict__ deg, int E) {
    int e = blockIdx.x * blockDim.x + threadIdx.x;
    if (e < E) atomicAdd(&deg[dst[e]], 1.0f);
}

// ---------------- dinv = rsqrt(deg) (deg >= 1 always) ----------------
__global__ void gcn_dinv(float* __restrict__ deg) {
    int i = blockIdx.x * blockDim.x + threadIdx.x;
    if (i < N_NODES) deg[i] = rsqrtf(deg[i]);
}

// ---------------- GEMM1: h1 = x(N x 3) @ W1(3 x 128), WMMA f32 16x16x4, K padded ----
// grid = 3125 blocks (50000/16), block = 256 (8 waves -> 8 col tiles of 16)
__global__ void gcn_gemm1(const float* __restrict__ x, const float* __restrict__ W1,
                          float* __restrict__ h1) {
    int lane = threadIdx.x & 31;
    int wave = threadIdx.x >> 5;          // col tile 0..7
    int rowBase = blockIdx.x * 16;
    int colBase = wave * 16;
    int m = lane & 15;
    int h = lane >> 4;                    // K-pair select: {0,1} or {2,3}
    int row = rowBase + m;
    int col = colBase + m;
    int k0 = 2 * h;
    int k1 = 2 * h + 1;

    v2f a, b;
    a[0] = x[row * INC + k0];                               // k0 in {0,2} < 3
    a[1] = (k1 < INC) ? x[row * INC + k1] : 0.0f;           // k=3 zero pad
    b[0] = W1[k0 * HID + col];
    b[1] = (k1 < INC) ? W1[k1 * HID + col] : 0.0f;

    v8f c = {};
    c = __builtin_amdgcn_wmma_f32_16x16x4_f32(false, a, false, b,
                                              (short)0, c, false, false);

    int outRow0 = rowBase + 8 * h;
#pragma unroll
    for (int i = 0; i < 8; ++i)
        h1[(outRow0 + i) * HID + col] = c[i];
}

// ---------------- edge scatter: accum[dst] += dinv[s]*dinv[d] * hsrc[src] ------
// one thread = one edge x 4 channels; shift = log2(C/4)
__global__ void gcn_scatter(const int* __restrict__ src, const int* __restrict__ dst,
                            const float* __restrict__ dinv, const float* __restrict__ hsrc,
                            float* __restrict__ accum, int E, int C, int shift) {
    int tid = blockIdx.x * blockDim.x + threadIdx.x;
    int e  = tid >> shift;
    int cg = tid & ((1 << shift) - 1);
    if (e >= E) return;
    int s = src[e];
    int d = dst[e];
    float nrm = dinv[s] * dinv[d];
    const float4* hp = (const float4*)(hsrc + (size_t)s * C) + cg;
    float4 v = *hp;
    float* ap = accum + (size_t)d * C + cg * 4;
    atomicAdd(ap + 0, nrm * v.x);
    atomicAdd(ap + 1, nrm * v.y);
    atomicAdd(ap + 2, nrm * v.z);
    atomicAdd(ap + 3, nrm * v.w);
}

// ---------------- layer-1 finalize: h1 = relu(agg1 + dinv^2*h1 + b1) in place --
__global__ void gcn_fin1(const float* __restrict__ agg1, float* __restrict__ h1,
                         const float* __restrict__ dinv, const float* __restrict__ b1) {
    int i = blockIdx.x * blockDim.x + threadIdx.x;
    if (i >= N_NODES * HID) return;
    int node = i >> 7;          // /128
    int cc   = i & (HID - 1);
    float di = dinv[node];
    float v = agg1[i] + di * di * h1[i] + b1[cc];
    h1[i] = fmaxf(v, 0.0f);
}

// ---------------- GEMM2: h2 = h1(N x 128) @ W2(128 x 64), 32x WMMA f32 16x16x4 -
// grid = 3125 blocks, block = 128 (4 waves -> 4 col tiles of 16)
__global__ void gcn_gemm2(const float* __restrict__ A, const float* __restrict__ W2,
                          float* __restrict__ h2) {
    int lane = threadIdx.x & 31;
    int wave = threadIdx.x >> 5;          // col tile 0..3
    int rowBase = blockIdx.x * 16;
    int colBase = wave * 16;
    int m = lane & 15;
    int h = lane >> 4;
    int row = rowBase + m;
    int col = colBase + m;

    const float* arow = A  + (size_t)row * HID + 2 * h;
    const float* bcol = W2 + (size_t)(2 * h) * OUTC + col;

    v8f c = {};
#pragma unroll 4
    for (int kk = 0; kk < HID / 4; ++kk) {
        v2f a, b;
        a[0] = arow[4 * kk + 0];
        a[1] = arow[4 * kk + 1];
        b[0] = bcol[(size_t)(4 * kk) * OUTC];
        b[1] = bcol[(size_t)(4 * kk + 1) * OUTC];
        c = __builtin_amdgcn_wmma_f32_16x16x4_f32(false, a, false, b,
                                                  (short)0, c, false, false);
    }

    int outRow0 = rowBase + 8 * h;
#pragma unroll
    for (int i = 0; i < 8; ++i)
        h2[(outRow0 + i) * OUTC + col] = c[i];
}

// ---------------- layer-2 finalize: out += dinv^2*h2 + b2 ----------------
__global__ void gcn_fin2(const float* __restrict__ h2, float* __restrict__ out,
                         const float* __restrict__ dinv, const float* __restrict__ b2) {
    int i = blockIdx.x * blockDim.x + threadIdx.x;
    if (i >= N_NODES * OUTC) return;
    int node = i >> 6;          // /64
    int cc   = i & (OUTC - 1);
    float di = dinv[node];
    out[i] += di * di * h2[i] + b2[cc];
}

extern "C" void kernel_launch(void* const* d_in, const int* in_sizes, int n_in,
                              void* d_out, int out_size, void* d_ws, size_t ws_size,
                              hipStream_t stream) {
    const float* x  = (const float*)d_in[0];
    const int*   ei = (const int*)d_in[1];
    const float* W1 = (const float*)d_in[2];
    const float* b1 = (const float*)d_in[3];
    const float* W2 = (const float*)d_in[4];
    const float* b2 = (const float*)d_in[5];
    float* out = (float*)d_out;

    const int E = in_sizes[1] / 2;
    const int* src = ei;
    const int* dst = ei + E;

    // workspace layout (floats): deg/dinv [N] | h1 [N*128] | agg1 [N*128]
    // h2 [N*64] aliases agg1 (agg1 is dead after gcn_fin1)
    float* deg  = (float*)d_ws;
    float* h1   = deg  + N_NODES;
    float* agg1 = h1   + (size_t)N_NODES * HID;
    float* h2   = agg1;   // alias: reuse after fin1 consumed agg1

    const int B = 256;

    // 1) init accumulators + self-loop degree
    gcn_init<<<(N_NODES * HID + B - 1) / B, B, 0, stream>>>(agg1, out, deg);
    // 2) in-degree counts
    gcn_degree<<<(E + B - 1) / B, B, 0, stream>>>(dst, deg, E);
    // 3) dinv = rsqrt(deg)
    gcn_dinv<<<(N_NODES + B - 1) / B, B, 0, stream>>>(deg);
    // 4) h1 = x @ W1   (WMMA, one 16x16 tile per wave)
    gcn_gemm1<<<N_NODES / 16, 256, 0, stream>>>(x, W1, h1);
    // 5) agg1[dst] += norm * h1[src]   (128 channels -> shift 5)
    gcn_scatter<<<(E * 32 + B - 1) / B, B, 0, stream>>>(src, dst, deg, h1, agg1, E, HID, 5);
    // 6) h1 = relu(agg1 + dinv^2*h1 + b1)
    gcn_fin1<<<(N_NODES * HID + B - 1) / B, B, 0, stream>>>(agg1, h1, deg, b1);
    // 7) h2 = h1 @ W2   (WMMA, 32 K-steps per tile)
    gcn_gemm2<<<N_NODES / 16, 128, 0, stream>>>(h1, W2, h2);
    // 8) out[dst] += norm * h2[src]    (64 channels -> shift 4)
    gcn_scatter<<<(E * 16 + B - 1) / B, B, 0, stream>>>(src, dst, deg, h2, out, E, OUTC, 4);
    // 9) out += dinv^2*h2 + b2
    gcn_fin2<<<(N_NODES * OUTC + B - 1) / B, B, 0, stream>>>(h2, out, deg, b2);
}